// PyramidROIAlign_5317169512505
// MI455X (gfx1250) — compile-verified
//
#include <hip/hip_runtime.h>
#include <stdint.h>

// PyramidROIAlign for MI455X (gfx1250).
// Bandwidth-bound gather op: one wave32 per (box, py) output row.
// 4 bilinear corner vectors (4 x 256 f32 = 4 KB) per pixel are staged to LDS
// with double-buffered GLOBAL_LOAD_ASYNC_TO_LDS_B128 (ASYNCcnt pipeline),
// combined with VALU FMAs, and streamed out with nontemporal b128 stores.

typedef float v4f __attribute__((ext_vector_type(4)));

#define POOLD 7
#define CCH   256
#define WPB   4   // waves per block (128 threads)

// Low 32 bits of a generic pointer into LDS == LDS byte address (ISA 10.2).
__device__ __forceinline__ unsigned lds_lo32(const void* p) {
  return (unsigned)(uintptr_t)p;
}

__device__ __forceinline__ void async_ld_b128(unsigned lds_addr, const float* gaddr) {
  // vdst = LDS byte address VGPR, addr = 64-bit global VGPR pair, saddr = off
  asm volatile("global_load_async_to_lds_b128 %0, %1, off"
               :: "v"(lds_addr), "v"(gaddr)
               : "memory");
}

__device__ __forceinline__ void wait_async_le8() {
  asm volatile("s_wait_asynccnt 8" ::: "memory");
}
__device__ __forceinline__ void wait_async_0() {
  asm volatile("s_wait_asynccnt 0" ::: "memory");
}

__global__ __launch_bounds__(WPB * 32)
void pyramid_roialign_kernel(const float* __restrict__ boxes,
                             const float* __restrict__ meta,
                             const float* __restrict__ p2,
                             const float* __restrict__ p3,
                             const float* __restrict__ p4,
                             const float* __restrict__ p5,
                             float* __restrict__ out,
                             int B, int N) {
  // [wave][double-buffer][4 corners * 256 ch] -> 4*2*1024 floats = 32 KB/block
  __shared__ __align__(16) float buf[WPB][2][4 * CCH];

  const int tid  = threadIdx.x;
  const int wave = tid >> 5;
  const int lane = tid & 31;
  const int unit = blockIdx.x * WPB + wave;       // (box, py) work unit
  const int total = B * N * POOLD;
  if (unit >= total) return;

  const int py  = unit % POOLD;
  const int box = unit / POOLD;                    // flat box index 0..B*N-1
  const int bimg = box / N;

  // ---- box + level selection (uniform across the wave) ----
  const float y1 = boxes[box * 4 + 0];
  const float x1 = boxes[box * 4 + 1];
  const float y2 = boxes[box * 4 + 2];
  const float x2 = boxes[box * 4 + 3];
  const float bh = y2 - y1;
  const float bw = x2 - x1;

  const float imh = meta[4];
  const float imw = meta[5];
  const float scale = 224.0f / sqrtf(imh * imw);           // 224/sqrt(area)
  const float lvlf  = log2f(sqrtf(fmaxf(bh * bw, 1e-30f)) / scale);
  int lvl = 4 + (int)rintf(lvlf);                           // round-half-even
  lvl = lvl < 2 ? 2 : (lvl > 5 ? 5 : lvl);

  const float* f;
  int H;
  if      (lvl == 2) { f = p2; H = 256; }
  else if (lvl == 3) { f = p3; H = 128; }
  else if (lvl == 4) { f = p4; H = 64;  }
  else               { f = p5; H = 32;  }
  const int W = H;
  f += (size_t)bimg * (size_t)H * (size_t)W * CCH;

  // ---- y interpolation for this output row ----
  const float fH1 = (float)(H - 1);
  const float gy  = (float)py * (1.0f / 6.0f);
  const float in_y = (y1 + gy * bh) * fH1;
  const bool  yv = (in_y >= 0.0f) && (in_y <= fH1);
  const float y0f = floorf(in_y);
  const int yi0 = (int)fminf(fmaxf(y0f, 0.0f), fH1);
  const int yi1 = (int)fminf(fmaxf(ceilf(in_y), 0.0f), fH1);
  const float wy = in_y - y0f;

  // ---- x interpolation for the 7 pixels (kept in registers via unroll) ----
  int   xi0[POOLD], xi1[POOLD];
  float wx[POOLD], vmask[POOLD];
  const float fW1 = (float)(W - 1);
#pragma unroll
  for (int px = 0; px < POOLD; ++px) {
    const float gx = (float)px * (1.0f / 6.0f);
    const float in_x = (x1 + gx * bw) * fW1;
    const bool xvb = (in_x >= 0.0f) && (in_x <= fW1);
    const float x0f = floorf(in_x);
    xi0[px] = (int)fminf(fmaxf(x0f, 0.0f), fW1);
    xi1[px] = (int)fminf(fmaxf(ceilf(in_x), 0.0f), fW1);
    wx[px]  = in_x - x0f;
    vmask[px] = (xvb && yv) ? 1.0f : 0.0f;
  }

  const unsigned lbase0 = lds_lo32(&buf[wave][0][0]);
  const unsigned lbase1 = lds_lo32(&buf[wave][1][0]);
  const size_t r0 = (size_t)yi0 * (size_t)W;
  const size_t r1 = (size_t)yi1 * (size_t)W;
  const size_t obox = (size_t)box * (POOLD * POOLD) * CCH + (size_t)py * POOLD * CCH;

  // stage pixel px's 4 corner vectors (4 KB) into LDS buffer pb: 8 async b128 ops
  auto stage = [&](int px, unsigned lbase) {
    const float* g[4];
    g[0] = f + (r0 + (size_t)xi0[px]) * CCH + lane * 4;
    g[1] = f + (r0 + (size_t)xi1[px]) * CCH + lane * 4;
    g[2] = f + (r1 + (size_t)xi0[px]) * CCH + lane * 4;
    g[3] = f + (r1 + (size_t)xi1[px]) * CCH + lane * 4;
#pragma unroll
    for (int k = 0; k < 4; ++k) {
      const unsigned l = lbase + (unsigned)(k * 1024) + (unsigned)(lane * 16);
      async_ld_b128(l,       g[k]);        // channels [lane*4 ..)
      async_ld_b128(l + 512, g[k] + 128);  // channels [128 + lane*4 ..)
    }
  };

  // bilinear combine from LDS + nontemporal streamed store
  auto compute = [&](int px, const float* lb) {
    const float vw = vmask[px];
    const float vx = wx[px];
    const float w00 = (1.0f - wy) * (1.0f - vx) * vw;
    const float w01 = (1.0f - wy) * vx * vw;
    const float w10 = wy * (1.0f - vx) * vw;
    const float w11 = wy * vx * vw;
    const size_t ob = obox + (size_t)px * CCH;
#pragma unroll
    for (int j = 0; j < 2; ++j) {
      const int c = j * 128 + lane * 4;
      const v4f a = *(const v4f*)(lb + 0 * CCH + c);   // ds_load_b128
      const v4f b = *(const v4f*)(lb + 1 * CCH + c);
      const v4f cc = *(const v4f*)(lb + 2 * CCH + c);
      const v4f d = *(const v4f*)(lb + 3 * CCH + c);
      const v4f r = a * w00 + b * w01 + cc * w10 + d * w11;
      __builtin_nontemporal_store(r, (v4f*)(out + ob + c));
    }
  };

  // ---- double-buffered async pipeline over the 7 pixels ----
  stage(0, lbase0);
#pragma unroll
  for (int px = 0; px < POOLD; ++px) {
    if (px < POOLD - 1) {
      stage(px + 1, ((px + 1) & 1) ? lbase1 : lbase0);
      wait_async_le8();   // drain previous batch (in-order completion)
    } else {
      wait_async_0();
    }
    compute(px, (px & 1) ? &buf[wave][1][0] : &buf[wave][0][0]);
  }
}

extern "C" void kernel_launch(void* const* d_in, const int* in_sizes, int n_in,
                              void* d_out, int out_size, void* d_ws, size_t ws_size,
                              hipStream_t stream) {
  const float* boxes = (const float*)d_in[0];
  const float* meta  = (const float*)d_in[1];
  const float* p2    = (const float*)d_in[2];
  const float* p3    = (const float*)d_in[3];
  const float* p4    = (const float*)d_in[4];
  const float* p5    = (const float*)d_in[5];
  float* out = (float*)d_out;

  // derive B, N from input sizes: p2 is B*256*256*256, boxes is B*N*4
  int B = in_sizes[2] / (256 * 256 * 256);
  if (B < 1) B = 1;
  int N = in_sizes[0] / (4 * B);

  const int total  = B * N * POOLD;           // one wave per (box, py)
  const int blocks = (total + WPB - 1) / WPB;

  pyramid_roialign_kernel<<<blocks, WPB * 32, 0, stream>>>(
      boxes, meta, p2, p3, p4, p5, out, B, N);
}